// Concat_84653805404637
// MI455X (gfx1250) — compile-verified
//
#include <hip/hip_runtime.h>

// Problem geometry (from reference): x is (40, 256, 256) f32.
// supports = x[0:10], queries = x[10:40].
// out[(q*10+s), t, 0:256]   = x[s, t, :]
// out[(q*10+s), t, 256:512] = x[10+q, t, :]
// out shape (300, 256, 512) f32, fully contiguous.
#define N_SUP_PAIRS 10
#define N_Q 30
#define T_DIM 256
#define F_DIM 256      // floats per input row
#define F4_DIM 64      // float4 per input row
#define OUT_F4 128     // float4 per output row
#define ROWS_PER_BLK 8 // t-rows per block -> 16 KB output per block

#if defined(__gfx1250__) &&                                                    \
    __has_builtin(__builtin_amdgcn_global_load_async_to_lds_b128) &&           \
    __has_builtin(__builtin_amdgcn_global_store_async_from_lds_b128) &&        \
    __has_builtin(__builtin_amdgcn_s_wait_asynccnt)
#define USE_ASYNC_LDS 1
#else
#define USE_ASYNC_LDS 0
#endif

typedef float __attribute__((ext_vector_type(4))) f4;

#if USE_ASYNC_LDS
typedef int __attribute__((ext_vector_type(4))) i4;
typedef __attribute__((address_space(1))) i4 i4_glb;
typedef __attribute__((address_space(3))) i4 i4_lds;
#endif

__global__ __launch_bounds__(256) void concat_bcast_kernel(
    const float* __restrict__ x, float* __restrict__ out) {
  const int tid = threadIdx.x;          // 0..255
  const int p   = blockIdx.y;           // 0..299  (= q*10 + s)
  const int q   = p / N_SUP_PAIRS;
  const int s   = p - q * N_SUP_PAIRS;
  const int t0  = blockIdx.x * ROWS_PER_BLK;

#if USE_ASYNC_LDS
  // LDS staged in final output layout for the 8-row chunk:
  // row r: [r*2048 .. r*2048+1023] = support half, [+1024 .. +2047] = query half
  __shared__ char smem[ROWS_PER_BLK * 2048];
  const int c = tid & 63;               // float4 column within input row
  auto* lds = (__attribute__((address_space(3))) char*)smem;

  // Memory -> LDS (async, 16B/lane): 8 KB support + 8 KB query per block.
#pragma unroll
  for (int j = 0; j < ROWS_PER_BLK / 4; ++j) {
    const int r = j * 4 + (tid >> 6);   // row within chunk
    const int t = t0 + r;
    const char* supg = (const char*)(x + (size_t)(s * T_DIM + t) * F_DIM + c * 4);
    const char* qryg = (const char*)(x + (size_t)((N_SUP_PAIRS + q) * T_DIM + t) * F_DIM + c * 4);
    __builtin_amdgcn_global_load_async_to_lds_b128(
        (i4_glb*)supg,
        (i4_lds*)(lds + (r * 2048 + c * 16)),
        0, 0);
    __builtin_amdgcn_global_load_async_to_lds_b128(
        (i4_glb*)qryg,
        (i4_lds*)(lds + (r * 2048 + 1024 + c * 16)),
        0, 0);
  }

  __builtin_amdgcn_s_wait_asynccnt(0);  // this wave's loads landed in LDS
  __syncthreads();                      // store regions cross wave boundaries

  // LDS -> Memory (async, 16B/lane): 16 KB contiguous output chunk.
  // cpol=1 -> TH=NT: write-once stream, don't sweep the L2 that holds inputs.
  char* outg = (char*)out + ((size_t)p * T_DIM + t0) * 2048;
#pragma unroll
  for (int k = 0; k < ROWS_PER_BLK / 2; ++k) {
    const int off = k * 4096 + tid * 16;
    __builtin_amdgcn_global_store_async_from_lds_b128(
        (i4_glb*)(outg + off),
        (i4_lds*)(lds + off),
        0, 1);
  }
  // S_ENDPGM performs an implicit wait-idle, covering outstanding async stores.
#else
  // Fallback: direct float4 copy, non-temporal stores.
  const f4* x4 = (const f4*)x;
  f4* out4 = (f4*)out;
  const size_t outBase =
      (size_t)p * (T_DIM * OUT_F4) + (size_t)blockIdx.x * (ROWS_PER_BLK * OUT_F4);
#pragma unroll
  for (int k = 0; k < ROWS_PER_BLK * OUT_F4 / 256; ++k) {
    const int idx = k * 256 + tid;      // float4 index within the chunk
    const int rr  = idx >> 7;           // row within chunk
    const int cc  = idx & 127;          // output column (float4)
    const int t   = t0 + rr;
    f4 v;
    if (cc < F4_DIM)
      v = x4[(size_t)(s * T_DIM + t) * F4_DIM + cc];
    else
      v = x4[(size_t)((N_SUP_PAIRS + q) * T_DIM + t) * F4_DIM + (cc - F4_DIM)];
    __builtin_nontemporal_store(v, &out4[outBase + idx]);
  }
#endif
}

extern "C" void kernel_launch(void* const* d_in, const int* in_sizes, int n_in,
                              void* d_out, int out_size, void* d_ws, size_t ws_size,
                              hipStream_t stream) {
  (void)in_sizes; (void)n_in; (void)out_size; (void)d_ws; (void)ws_size;
  const float* x = (const float*)d_in[0];
  float* out = (float*)d_out;
  dim3 grid(T_DIM / ROWS_PER_BLK, N_Q * N_SUP_PAIRS);  // (32, 300)
  dim3 block(256);
  concat_bcast_kernel<<<grid, block, 0, stream>>>(x, out);
}